// FlowEmbed_54631984005448
// MI455X (gfx1250) — compile-verified
//
#include <hip/hip_runtime.h>
#include <math.h>

#define BQ   2
#define NQ   4096
#define KQ   32
#define DIMQ 128
#define RAD2 (0.2f * 0.2f)

typedef float v2f __attribute__((ext_vector_type(2)));
typedef float v8f __attribute__((ext_vector_type(8)));

// ------------------------------------------------------------------
// Kernel 1: ball query (x,y only; z zeroed in reference).
// One thread per query point; point tile staged in LDS.
// First-K-in-index-order semantics match argsort(keyv)[:K]; pad with self.
// ------------------------------------------------------------------
__global__ __launch_bounds__(256) void fe_ball_query(
    const float* __restrict__ xyzp, int* __restrict__ idx)
{
  const int b = blockIdx.y;
  const int n = blockIdx.x * 256 + threadIdx.x;
  const float* base = xyzp + (size_t)b * NQ * 4;
  const float px = base[n * 4 + 0];
  const float py = base[n * 4 + 1];
  __shared__ float sxl[256], syl[256];
  int cnt = 0;
  int* out = idx + ((size_t)b * NQ + n) * KQ;
  for (int m0 = 0; m0 < NQ; m0 += 256) {
    float vx = base[(m0 + threadIdx.x) * 4 + 0];
    float vy = base[(m0 + threadIdx.x) * 4 + 1];
    __syncthreads();
    sxl[threadIdx.x] = vx;
    syl[threadIdx.x] = vy;
    __syncthreads();
    if (cnt < KQ) {
      for (int j = 0; j < 256; ++j) {
        float dx = px - sxl[j], dy = py - syl[j];
        if (dx * dx + dy * dy < RAD2 && cnt < KQ) out[cnt++] = m0 + j;
      }
    }
  }
  for (; cnt < KQ; ++cnt) out[cnt] = n;
}

// ------------------------------------------------------------------
// Kernel 2: g0 = F @ Wm[0:128], g1 = F @ Wm[128:256]  (fp32 WMMA 16x16x4)
// One wave per 16x16 output tile; 8 waves per block.
// A frag (16x4): lanes 0-15 hold row M=l, K=kk+0/kk+1; lanes 16-31 K=kk+2/kk+3.
// B frag (4x16): v0 = W[kk+2*half, col], v1 = W[kk+1+2*half, col].
// C/D (16x16 f32): vgpr r -> M = r + 8*(lane>=16), N = lane&15.
// ------------------------------------------------------------------
__global__ __launch_bounds__(256) void fe_gemm_g01(
    const float* __restrict__ X, const float* __restrict__ Wm,
    float* __restrict__ g0, float* __restrict__ g1)
{
#if __has_builtin(__builtin_amdgcn_wmma_f32_16x16x4_f32)
  const int lane = threadIdx.x & 31;
  const int wave = threadIdx.x >> 5;
  const int t    = blockIdx.x * 8 + wave;     // 0..8191 tiles
  const int sel  = t >> 12;                   // 0: g0, 1: g1
  const int rem  = t & 4095;
  const int mt   = rem >> 3;                  // 0..511 (M tiles over 8192 rows)
  const int nt   = rem & 7;                   // 0..7   (N tiles over 128 cols)
  const float* Wb = Wm + (size_t)sel * DIMQ * DIMQ;
  float* G = sel ? g1 : g0;
  const int half = lane >> 4;
  const int l15  = lane & 15;
  const int arow = mt * 16 + l15;
  const int bcol = nt * 16 + l15;
  v8f acc = {};
  for (int kk = 0; kk < DIMQ; kk += 4) {
    const int ka = kk + 2 * half;
    v2f a = *(const v2f*)(X + (size_t)arow * DIMQ + ka);
    v2f bf;
    bf.x = Wb[(size_t)ka * DIMQ + bcol];
    bf.y = Wb[(size_t)(ka + 1) * DIMQ + bcol];
    acc = __builtin_amdgcn_wmma_f32_16x16x4_f32(
        false, a, false, bf, (short)0, acc, false, false);
  }
#pragma unroll
  for (int r = 0; r < 8; ++r) {
    G[(size_t)(mt * 16 + r + 8 * half) * DIMQ + bcol] = acc[r];
  }
#else
  // exact fp32 fallback (keeps compile green if builtin is absent)
  const int tot = gridDim.x * 256;
  for (int o = blockIdx.x * 256 + threadIdx.x; o < 2 * BQ * NQ * DIMQ; o += tot) {
    int sel = o / (BQ * NQ * DIMQ);
    int r   = o % (BQ * NQ * DIMQ);
    int row = r / DIMQ, c = r % DIMQ;
    const float* Wb = Wm + (size_t)sel * DIMQ * DIMQ;
    float s = 0.f;
    for (int j = 0; j < DIMQ; ++j)
      s = fmaf(X[(size_t)row * DIMQ + j], Wb[(size_t)j * DIMQ + c], s);
    (sel ? g1 : g0)[r] = s;
  }
#endif
}

// ------------------------------------------------------------------
// Kernel 3: per-block partial sums of speed (for mean_speed), deterministic.
// ------------------------------------------------------------------
__global__ __launch_bounds__(256) void fe_speed_partial(
    const float* __restrict__ xyzp, const int* __restrict__ idx,
    float2* __restrict__ partial)
{
  const int b = blockIdx.y;
  const int n = blockIdx.x * 256 + threadIdx.x;
  const float* base = xyzp + (size_t)b * NQ * 4;
  float4 q = *(const float4*)(base + (size_t)n * 4);
  const int* id = idx + ((size_t)b * NQ + n) * KQ;
  float sx = 0.f, sy = 0.f;
  for (int k = 0; k < KQ; ++k) {
    int m = id[k];
    float4 pm = *(const float4*)(base + (size_t)m * 4);
    float dx = q.x - pm.x, dy = q.y - pm.y, dz = q.z - pm.z;
    float ts = (dz == 0.0f) ? 1e-6f : dz;
    sx += dx / ts * 640.0f;
    sy += dy / ts * 480.0f;
  }
  __shared__ float rx[256], ry[256];
  const int tid = threadIdx.x;
  rx[tid] = sx; ry[tid] = sy;
  __syncthreads();
  for (int s = 128; s > 0; s >>= 1) {
    if (tid < s) { rx[tid] += rx[tid + s]; ry[tid] += ry[tid + s]; }
    __syncthreads();
  }
  if (tid == 0) partial[b * gridDim.x + blockIdx.x] = make_float2(rx[0], ry[0]);
}

// ------------------------------------------------------------------
// Kernel 4: fold mean_speed @ Wm[263:265] + b into per-batch bias vector.
// ------------------------------------------------------------------
__global__ __launch_bounds__(128) void fe_bias_build(
    const float2* __restrict__ partial, int nPart,
    const float* __restrict__ Wm, const float* __restrict__ bvec,
    float* __restrict__ biasT)
{
  const int b = blockIdx.x;
  const int c = threadIdx.x;
  __shared__ float msx, msy;
  if (c == 0) {
    float sx = 0.f, sy = 0.f;
    for (int i = 0; i < nPart; ++i) { sx += partial[b * nPart + i].x; sy += partial[b * nPart + i].y; }
    const float inv = 1.0f / ((float)NQ * (float)KQ);
    msx = sx * inv; msy = sy * inv;
  }
  __syncthreads();
  biasT[b * DIMQ + c] = bvec[c] + msx * Wm[263 * DIMQ + c] + msy * Wm[264 * DIMQ + c];
}

// ------------------------------------------------------------------
// Kernel 5 (pass A): h[n,k,c] = g0[n,c] + g1[idx,c] + geo7·W2 + bias.
// One block per (b,n), one thread per channel. Emits hmax/hmin/Σh/Σh².
// ------------------------------------------------------------------
__global__ __launch_bounds__(128) void fe_passA(
    const float* __restrict__ xyzp, const int* __restrict__ idx,
    const float* __restrict__ g0, const float* __restrict__ g1,
    const float* __restrict__ Wm, const float* __restrict__ biasT,
    float* __restrict__ hmaxA, float* __restrict__ hminA,
    float* __restrict__ sArr, float* __restrict__ ssArr)
{
  const int bn = blockIdx.x;
  const int b  = bn >> 12;        // N = 4096
  const int n  = bn & 4095;
  const int c  = threadIdx.x;
  __shared__ float geo[KQ][7];
  __shared__ int   sm[KQ];
  const float* base = xyzp + (size_t)b * NQ * 4;
  if (c < KQ) {
    int m = idx[(size_t)bn * KQ + c];
    sm[c] = m;
    float4 q  = *(const float4*)(base + (size_t)n * 4);
    float4 pm = *(const float4*)(base + (size_t)m * 4);
    float dx = q.x - pm.x, dy = q.y - pm.y, dz = q.z - pm.z, dw = q.w - pm.w;
    float ts = (dz == 0.0f) ? 1e-6f : dz;
    geo[c][0] = dx; geo[c][1] = dy; geo[c][2] = dz; geo[c][3] = dw;
    geo[c][4] = dx / ts * 640.0f;
    geo[c][5] = dy / ts * 480.0f;
    geo[c][6] = fabsf(dw);
  }
  __syncthreads();
  float w2[7];
#pragma unroll
  for (int j = 0; j < 7; ++j) w2[j] = Wm[(size_t)(256 + j) * DIMQ + c];
  const float basev = g0[(size_t)bn * DIMQ + c] + biasT[b * DIMQ + c];
  const float* g1b = g1 + (size_t)b * NQ * DIMQ;
  float hmx = -3.4e38f, hmn = 3.4e38f, s = 0.f, ss = 0.f;
  for (int k = 0; k < KQ; ++k) {
    int m = sm[k];
    float h = basev + g1b[(size_t)m * DIMQ + c];
#pragma unroll
    for (int j = 0; j < 7; ++j) h = fmaf(geo[k][j], w2[j], h);
    hmx = fmaxf(hmx, h);
    hmn = fminf(hmn, h);
    s += h;
    ss = fmaf(h, h, ss);
  }
  const size_t o = (size_t)bn * DIMQ + c;
  hmaxA[o] = hmx; hminA[o] = hmn; sArr[o] = s; ssArr[o] = ss;
}

// ------------------------------------------------------------------
// Kernel 6: per-channel mu/var over all (B,N,K); fixed-order reduction.
// Emits (a, c2) with a = gamma/sqrt(var+eps), c2 = beta - mu*a.
// ------------------------------------------------------------------
__global__ __launch_bounds__(256) void fe_stats(
    const float* __restrict__ sArr, const float* __restrict__ ssArr,
    const float* __restrict__ gamma, const float* __restrict__ beta,
    float2* __restrict__ sc)
{
  const int c = blockIdx.x;          // 128 channels
  const int tid = threadIdx.x;
  const int BN = BQ * NQ;
  float s = 0.f, ss = 0.f;
  for (int r = tid; r < BN; r += 256) {
    s  += sArr[(size_t)r * DIMQ + c];
    ss += ssArr[(size_t)r * DIMQ + c];
  }
  __shared__ float a1[256], a2[256];
  a1[tid] = s; a2[tid] = ss;
  __syncthreads();
  for (int st = 128; st > 0; st >>= 1) {
    if (tid < st) { a1[tid] += a1[tid + st]; a2[tid] += a2[tid + st]; }
    __syncthreads();
  }
  if (tid == 0) {
    const float cntf = (float)BN * (float)KQ;
    float mu  = a1[0] / cntf;
    float var = a2[0] / cntf - mu * mu;
    float a   = gamma[c] * rsqrtf(var + 1e-5f);
    sc[c] = make_float2(a, beta[c] - mu * a);
  }
}

// ------------------------------------------------------------------
// Kernel 7: out = max_k relu(a*h + c2) = max(max(a*hmax+c2, a*hmin+c2), 0)
// (exact: affine-per-channel + monotone ReLU commute with max over k)
// ------------------------------------------------------------------
__global__ __launch_bounds__(256) void fe_out(
    const float* __restrict__ hmaxA, const float* __restrict__ hminA,
    const float2* __restrict__ sc, float* __restrict__ out)
{
  const int i = blockIdx.x * 256 + threadIdx.x;
  const int c = i & (DIMQ - 1);
  float2 p = sc[c];
  float v1 = fmaf(p.x, hmaxA[i], p.y);
  float v2 = fmaf(p.x, hminA[i], p.y);
  out[i] = fmaxf(fmaxf(v1, v2), 0.0f);
}

// ------------------------------------------------------------------
extern "C" void kernel_launch(void* const* d_in, const int* in_sizes, int n_in,
                              void* d_out, int out_size, void* d_ws, size_t ws_size,
                              hipStream_t stream)
{
  (void)in_sizes; (void)n_in; (void)out_size; (void)ws_size;
  const float* xyzp  = (const float*)d_in[0];
  const float* feat  = (const float*)d_in[1];
  const float* Wm    = (const float*)d_in[2];
  const float* bvec  = (const float*)d_in[3];
  const float* gamma = (const float*)d_in[4];
  const float* beta  = (const float*)d_in[5];
  float* out = (float*)d_out;

  char* ws = (char*)d_ws;
  size_t off = 0;
  auto take = [&](size_t bytes) -> char* {
    char* p = ws + off;
    off += (bytes + 255) & ~(size_t)255;
    return p;
  };
  int*    idx   = (int*)   take((size_t)BQ * NQ * KQ * sizeof(int));       // 1 MB
  float*  g0    = (float*) take((size_t)BQ * NQ * DIMQ * sizeof(float));   // 4 MB
  float*  g1    = (float*) take((size_t)BQ * NQ * DIMQ * sizeof(float));   // 4 MB
  float*  hmaxA = (float*) take((size_t)BQ * NQ * DIMQ * sizeof(float));   // 4 MB
  float*  hminA = (float*) take((size_t)BQ * NQ * DIMQ * sizeof(float));   // 4 MB
  float*  sArr  = (float*) take((size_t)BQ * NQ * DIMQ * sizeof(float));   // 4 MB
  float*  ssArr = (float*) take((size_t)BQ * NQ * DIMQ * sizeof(float));   // 4 MB
  float*  biasT = (float*) take((size_t)BQ * DIMQ * sizeof(float));
  float2* part  = (float2*)take((size_t)BQ * (NQ / 256) * sizeof(float2));
  float2* sc    = (float2*)take((size_t)DIMQ * sizeof(float2));

  fe_ball_query   <<<dim3(NQ / 256, BQ), 256, 0, stream>>>(xyzp, idx);
  fe_gemm_g01     <<<dim3(1024),         256, 0, stream>>>(feat, Wm, g0, g1);
  fe_speed_partial<<<dim3(NQ / 256, BQ), 256, 0, stream>>>(xyzp, idx, part);
  fe_bias_build   <<<dim3(BQ),           128, 0, stream>>>(part, NQ / 256, Wm, bvec, biasT);
  fe_passA        <<<dim3(BQ * NQ),      128, 0, stream>>>(xyzp, idx, g0, g1, Wm, biasT,
                                                           hmaxA, hminA, sArr, ssArr);
  fe_stats        <<<dim3(DIMQ),         256, 0, stream>>>(sArr, ssArr, gamma, beta, sc);
  fe_out          <<<dim3((BQ * NQ * DIMQ) / 256), 256, 0, stream>>>(hmaxA, hminA, sc, out);
}